// RBF_6073083756915
// MI455X (gfx1250) — compile-verified
//
#include <hip/hip_runtime.h>

typedef float v2f __attribute__((ext_vector_type(2)));
typedef float v4f __attribute__((ext_vector_type(4)));
typedef float v8f __attribute__((ext_vector_type(8)));

#define BATCH 64
#define NPTS  64
#define NMU   300
#define DMU   (30.0f / 299.0f)
#define DIAG_FILL 1000.0f

// ---------------------------------------------------------------------------
// Phase 1: pairwise distances via V_WMMA_F32_16X16X4_F32.
// One wave32 per 16x16 tile of the 64x64 distance matrix, 64 batches.
// A is pre-scaled by -2 so D = -2 * x_i . x_j, then r^2 = |x_i|^2 + |x_j|^2 + D.
// A-layout (32-bit 16x4): lanes 0-15 -> {K0,K1}, lanes 16-31 -> {K2,K3}; K3 = 0.
// B-layout (32-bit 4x16): mirror, N = lane&15.
// D-layout (f32 16x16):   VGPR v, lane L -> (M = v + 8*(L>>4), N = L&15).
// No divergence anywhere: diagonal handled with a value select (v_cndmask),
// sqrt via raw v_sqrt_f32 so clang if-converts instead of branching.
// ---------------------------------------------------------------------------
__global__ __launch_bounds__(128) void rbf_dist_wmma(const float* __restrict__ coords,
                                                     float* __restrict__ r_ws) {
    const int lane = threadIdx.x & 31;
    const int wave = threadIdx.x >> 5;
    const int tile = blockIdx.x * 4 + wave;      // 1024 tiles total
    const int b    = tile >> 4;
    const int ti   = (tile >> 2) & 3;
    const int tj   = tile & 3;

    const int n  = lane & 15;
    const int hi = lane >> 4;

    const float* cb = coords + b * NPTS * 3;

    const int rowA = ti * 16 + n;   // A: M = lane&15
    const int rowB = tj * 16 + n;   // B: N = lane&15

    v2f a, bm;
    a.x  = -2.0f * cb[rowA * 3 + (hi ? 2 : 0)];
    a.y  = hi ? 0.0f : (-2.0f * cb[rowA * 3 + 1]);
    bm.x = cb[rowB * 3 + (hi ? 2 : 0)];
    bm.y = hi ? 0.0f : cb[rowB * 3 + 1];

    v8f c = {0.f, 0.f, 0.f, 0.f, 0.f, 0.f, 0.f, 0.f};
    // 8 args: (neg_a, A, neg_b, B, c_mod, C, reuse_a, reuse_b)
    v8f d = __builtin_amdgcn_wmma_f32_16x16x4_f32(false, a, false, bm,
                                                  (short)0, c, false, false);

    // column norm |x_j|^2 (same for all 8 accumulator rows this lane holds)
    const float xj = cb[rowB * 3 + 0];
    const float yj = cb[rowB * 3 + 1];
    const float zj = cb[rowB * 3 + 2];
    const float nj = xj * xj + yj * yj + zj * zj;

    const int jg = tj * 16 + n;
#pragma unroll
    for (int v = 0; v < 8; ++v) {
        const int ig = ti * 16 + v + 8 * hi;
        const float xi = cb[ig * 3 + 0];
        const float yi = cb[ig * 3 + 1];
        const float zi = cb[ig * 3 + 2];
        const float ni = xi * xi + yi * yi + zi * zi;
        const float r2 = ni + nj + d[v];
        const float rr = __builtin_amdgcn_sqrtf(fmaxf(r2, 0.0f)); // raw v_sqrt_f32
        const float r  = (ig == jg) ? DIAG_FILL : rr;             // v_cndmask, no branch
        r_ws[(b * NPTS + ig) * NPTS + jg] = r;
    }
}

// ---------------------------------------------------------------------------
// Phase 2: RBF expansion, the bandwidth kernel (314.6 MB of stores).
// One thread per 4 consecutive mu values -> coalesced 128-bit NT stores.
// 75 consecutive threads share one r value (L0/L2 hit on the 1 MB r table).
// ---------------------------------------------------------------------------
__global__ __launch_bounds__(256) void rbf_expand(const float* __restrict__ r_ws,
                                                  float* __restrict__ out) {
    const unsigned q    = blockIdx.x * 256u + threadIdx.x;  // quad index
    const unsigned pair = q / 75u;                          // (b,i,j) flat index
    const unsigned muq  = q - pair * 75u;                   // which group of 4 mu

    const float r = r_ws[pair];

    v4f v;
#pragma unroll
    for (int e = 0; e < 4; ++e) {
        const float mu = (float)(muq * 4u + e) * DMU;
        const float t  = r - mu;
        v[e] = __expf(-10.0f * t * t);                      // v_mul + v_exp_f32
    }

    __builtin_nontemporal_store(v, (v4f*)out + q);          // global_store_b128 NT
}

extern "C" void kernel_launch(void* const* d_in, const int* in_sizes, int n_in,
                              void* d_out, int out_size, void* d_ws, size_t ws_size,
                              hipStream_t stream) {
    const float* coords = (const float*)d_in[0];
    float*       out    = (float*)d_out;
    float*       r_ws   = (float*)d_ws;   // 64*64*64 floats = 1 MB scratch

    // Phase 1: 1024 wave-tiles, 4 waves per 128-thread block -> 256 blocks.
    rbf_dist_wmma<<<dim3(256), dim3(128), 0, stream>>>(coords, r_ws);

    // Phase 2: 78,643,200 outputs / 4 per thread = 19,660,800 threads.
    rbf_expand<<<dim3(76800), dim3(256), 0, stream>>>(r_ws, out);
}